// NeighborSearch_60026462929315
// MI455X (gfx1250) — compile-verified
//
#include <hip/hip_runtime.h>

// Radius neighbor search, CDNA5 (gfx1250, wave32).
// Passes:
//  1) count_kernel : V_WMMA_F32_16X16X4_F32 tiles -> per-query neighbor counts
//  2) scan_kernel  : single-block exclusive scan -> row_splits (in d_out tail)
//  3) fill_kernel  : -1 fill of neighbors_index via B128 stores (HBM-bound, dominant cost)
//  4) emit_kernel  : WMMA tiles + wave32 ballot prefix-popcount compaction -> packed indices

typedef float v2f __attribute__((ext_vector_type(2)));
typedef float v8f __attribute__((ext_vector_type(8)));

__device__ __forceinline__ unsigned ballot32(bool p) {
    // wave32: mask lives in the low 32 bits
    return (unsigned)__ballot(p ? 1 : 0);
}

// ---------------------------------------------------------------------------
// Pass 1: per-query neighbor counts. One wave handles a 16-query tile vs all n.
// ---------------------------------------------------------------------------
__global__ __launch_bounds__(256) void nns_count_kernel(
    const float* __restrict__ Q, const float* __restrict__ D,
    const float* __restrict__ radius_p, int* __restrict__ counts, int n) {
    const int lane = threadIdx.x & 31;
    const int wid  = threadIdx.x >> 5;
    const int qtile = blockIdx.x * (blockDim.x >> 5) + wid;
    const int qbase = qtile * 16;
    const int hi  = lane >> 4;    // 0: lanes 0-15 (K=0,1) | 1: lanes 16-31 (K=2,3)
    const int sub = lane & 15;
    const float r  = radius_p[0];
    const float r2 = r * r;

    // A tile (16x4): lane's query row m = qbase+sub. v0 = K0|K2, v1 = K1|K3(pad 0)
    const float* q = Q + (size_t)(qbase + sub) * 3;
    v2f a;
    a.x = hi ? q[2] : q[0];
    a.y = hi ? 0.0f : q[1];

    // ||q||^2 for this lane's 8 accumulator rows: M = j + 8*hi
    float q2row[8];
#pragma unroll
    for (int j = 0; j < 8; ++j) {
        const float* qq = Q + (size_t)(qbase + j + 8 * hi) * 3;
        q2row[j] = qq[0] * qq[0] + qq[1] * qq[1] + qq[2] * qq[2];
    }

    int cnt[8];
#pragma unroll
    for (int j = 0; j < 8; ++j) cnt[j] = 0;

    for (int nbase = 0; nbase < n; nbase += 16) {
        // B tile (4x16): lane's data column = nbase+sub (same for both halves)
        const float* d = D + (size_t)(nbase + sub) * 3;
        float dx = d[0], dy = d[1], dz = d[2];
        v2f b;
        b.x = hi ? dz : dx;
        b.y = hi ? 0.0f : dy;
        float d2 = dx * dx + dy * dy + dz * dz;

        v8f c = {};
        c = __builtin_amdgcn_wmma_f32_16x16x4_f32(
            /*neg_a=*/false, a, /*neg_b=*/false, b,
            /*c_mod=*/(short)0, c, /*reuse_a=*/false, /*reuse_b=*/false);

#pragma unroll
        for (int j = 0; j < 8; ++j) {
            float dist2 = q2row[j] + d2 - 2.0f * c[j];
            unsigned m = ballot32(dist2 <= r2);
            // low 16 bits = row j, high 16 bits = row j+8
            cnt[j] += __popc(hi ? (m >> 16) : (m & 0xFFFFu));
        }
    }
    // one writer per row: lane with sub==j in each half owns row j+8*hi
#pragma unroll
    for (int j = 0; j < 8; ++j)
        if (sub == j) counts[qbase + j + 8 * hi] = cnt[j];
}

// ---------------------------------------------------------------------------
// Pass 2: exclusive scan of counts -> row_splits[0..m], in place (io aliases).
// Single block of 1024 threads, m assumed divisible by 1024 (8192 here).
// ---------------------------------------------------------------------------
__global__ __launch_bounds__(1024) void nns_scan_kernel(int* __restrict__ io, int m) {
    __shared__ int s[1024];
    const int tid = threadIdx.x;
    const int seg = m >> 10;  // 8 for m=8192
    int loc[8];
    int sum = 0;
#pragma unroll
    for (int i = 0; i < 8; ++i) {
        if (i < seg) {
            loc[i] = sum;
            sum += io[tid * seg + i];
        }
    }
    s[tid] = sum;
    __syncthreads();
    // Hillis-Steele inclusive scan (reads complete before writes each step)
    for (int off = 1; off < 1024; off <<= 1) {
        int v = (tid >= off) ? s[tid - off] : 0;
        __syncthreads();
        s[tid] += v;
        __syncthreads();
    }
    int excl = s[tid] - sum;
#pragma unroll
    for (int i = 0; i < 8; ++i)
        if (i < seg) io[tid * seg + i] = excl + loc[i];
    if (tid == 1023) io[m] = s[1023];
}

// ---------------------------------------------------------------------------
// Pass 3: fill neighbors_index with -1 using 128-bit stores (dominant HBM cost).
// ---------------------------------------------------------------------------
__global__ __launch_bounds__(256) void nns_fill_kernel(int4* __restrict__ out,
                                                       long long nvec) {
    long long i = (long long)blockIdx.x * blockDim.x + threadIdx.x;
    long long stride = (long long)gridDim.x * blockDim.x;
    const int4 v = make_int4(-1, -1, -1, -1);
    for (; i < nvec; i += stride) out[i] = v;
}

// ---------------------------------------------------------------------------
// Pass 4: recompute tiles with WMMA, compact passing column indices to
// out[row_splits[row] + running_offset] via ballot prefix-popcount.
// ---------------------------------------------------------------------------
__global__ __launch_bounds__(256) void nns_emit_kernel(
    const float* __restrict__ Q, const float* __restrict__ D,
    const float* __restrict__ radius_p, const int* __restrict__ splits,
    int* __restrict__ out, int n) {
    const int lane = threadIdx.x & 31;
    const int wid  = threadIdx.x >> 5;
    const int qtile = blockIdx.x * (blockDim.x >> 5) + wid;
    const int qbase = qtile * 16;
    const int hi  = lane >> 4;
    const int sub = lane & 15;
    const float r  = radius_p[0];
    const float r2 = r * r;

    const float* q = Q + (size_t)(qbase + sub) * 3;
    v2f a;
    a.x = hi ? q[2] : q[0];
    a.y = hi ? 0.0f : q[1];

    float q2row[8];
    int base[8], cur[8];
#pragma unroll
    for (int j = 0; j < 8; ++j) {
        const int row = qbase + j + 8 * hi;
        const float* qq = Q + (size_t)row * 3;
        q2row[j] = qq[0] * qq[0] + qq[1] * qq[1] + qq[2] * qq[2];
        base[j] = splits[row];
        cur[j]  = 0;
    }

    const unsigned lt = (1u << sub) - 1u;  // lanes-below-me mask within half

    for (int nbase = 0; nbase < n; nbase += 16) {
        const float* d = D + (size_t)(nbase + sub) * 3;
        float dx = d[0], dy = d[1], dz = d[2];
        v2f b;
        b.x = hi ? dz : dx;
        b.y = hi ? 0.0f : dy;
        float d2 = dx * dx + dy * dy + dz * dz;

        v8f c = {};
        c = __builtin_amdgcn_wmma_f32_16x16x4_f32(
            false, a, false, b, (short)0, c, false, false);

#pragma unroll
        for (int j = 0; j < 8; ++j) {
            float dist2 = q2row[j] + d2 - 2.0f * c[j];
            bool p = dist2 <= r2;
            unsigned m = ballot32(p);
            unsigned half = hi ? (m >> 16) : (m & 0xFFFFu);
            if (p) {
                int pos = base[j] + cur[j] + __popc(half & lt);
                out[pos] = nbase + sub;  // ascending column order preserved
            }
            cur[j] += __popc(half);
        }
    }
}

// ---------------------------------------------------------------------------
extern "C" void kernel_launch(void* const* d_in, const int* in_sizes, int n_in,
                              void* d_out, int out_size, void* d_ws, size_t ws_size,
                              hipStream_t stream) {
    (void)n_in; (void)out_size; (void)d_ws; (void)ws_size;
    const float* data    = (const float*)d_in[0];
    const float* queries = (const float*)d_in[1];
    const float* radius  = (const float*)d_in[2];
    const int n = in_sizes[0] / 3;  // 8192 data points
    const int m = in_sizes[1] / 3;  // 8192 queries

    int* out = (int*)d_out;
    const long long mn = (long long)m * (long long)n;
    int* splits = out + mn;  // row_splits region of d_out (m+1 ints)

    // 8 waves/block, each wave owns one 16-query tile
    const int wavesPerBlock = 8;
    const int qtiles = m / 16;
    const int tileBlocks = qtiles / wavesPerBlock;  // 64 for m=8192

    // 1) counts -> stored into the row_splits region (scanned in place next)
    nns_count_kernel<<<tileBlocks, 32 * wavesPerBlock, 0, stream>>>(
        queries, data, radius, splits, n);

    // 2) in-place exclusive scan -> row_splits
    nns_scan_kernel<<<1, 1024, 0, stream>>>(splits, m);

    // 3) -1 fill of neighbors_index (mn divisible by 4; 128-bit stores)
    nns_fill_kernel<<<4096, 256, 0, stream>>>((int4*)out, mn / 4);

    // 4) packed index emission (overwrites the front of the filled region)
    nns_emit_kernel<<<tileBlocks, 32 * wavesPerBlock, 0, stream>>>(
        queries, data, radius, splits, out, n);
}